// LigandPocketPoolingHead_60438779789616
// MI455X (gfx1250) — compile-verified
//
#include <hip/hip_runtime.h>
#include <hip/hip_bf16.h>
#include <math.h>

// ---- problem constants (fixed by the reference) ----
#define N_NODES 150000
#define NG      512      // num_graphs
#define CH      192      // c_hidden
#define CDN     32       // c_hidden / 6
#define FEAT1   393      // 12*CDN + 9  (per-entry feature width)
#define KPAD    800      // 2*FEAT1 = 786 padded up to a multiple of 16
#define STRIP   8        // nodes per thread in segment-sum kernels

typedef float v2f __attribute__((ext_vector_type(2)));
typedef float v8f __attribute__((ext_vector_type(8)));

__device__ __forceinline__ float silu_f(float x) { return x * (1.0f / (1.0f + __expf(-x))); }

// ---------------------------------------------------------------------------
// Segment sum of per-node 3-vectors (+count) into comcnt[G][4].
// batch is sorted, so each thread accumulates a strip and flushes atomics
// only on graph-boundary crossings (~1 flush per strip).
// useIC=0: v = coords[n];  useIC=1: v = icv[n] * ncv[n]
// ---------------------------------------------------------------------------
__global__ void seg_sum_vec3_kernel(const float* __restrict__ coords,
                                    const float* __restrict__ icv,
                                    const float* __restrict__ ncv,
                                    const int* __restrict__ batch,
                                    float* __restrict__ comcnt, int useIC) {
  int t = blockIdx.x * blockDim.x + threadIdx.x;
  int base = t * STRIP;
  if (base >= N_NODES) return;
  int end = base + STRIP; if (end > N_NODES) end = N_NODES;
  float sx = 0.f, sy = 0.f, sz = 0.f, sc = 0.f;
  int cur = batch[base];
  for (int n = base; n < end; ++n) {
    int b = batch[n];
    if (b != cur) {
      atomicAdd(&comcnt[cur * 4 + 0], sx); atomicAdd(&comcnt[cur * 4 + 1], sy);
      atomicAdd(&comcnt[cur * 4 + 2], sz); atomicAdd(&comcnt[cur * 4 + 3], sc);
      sx = sy = sz = sc = 0.f; cur = b;
    }
    float x, y, z;
    if (useIC) { float s = icv[n]; x = s * ncv[n * 3 + 0]; y = s * ncv[n * 3 + 1]; z = s * ncv[n * 3 + 2]; }
    else       { x = coords[n * 3 + 0]; y = coords[n * 3 + 1]; z = coords[n * 3 + 2]; }
    sx += x; sy += y; sz += z; sc += 1.0f;
  }
  atomicAdd(&comcnt[cur * 4 + 0], sx); atomicAdd(&comcnt[cur * 4 + 1], sy);
  atomicAdd(&comcnt[cur * 4 + 2], sz); atomicAdd(&comcnt[cur * 4 + 3], sc);
}

__global__ void finalize_com_kernel(float* __restrict__ comcnt) {
  int g = blockIdx.x * blockDim.x + threadIdx.x;
  if (g >= NG) return;
  float inv = 1.0f / fmaxf(comcnt[g * 4 + 3], 1.0f);
  comcnt[g * 4 + 0] *= inv; comcnt[g * 4 + 1] *= inv; comcnt[g * 4 + 2] *= inv;
}

// ---------------------------------------------------------------------------
// Center by COM, write normalized directions, accumulate 3x3 covariance
// (6 unique entries) per graph with strip-flushed atomics.
// ---------------------------------------------------------------------------
__global__ void center_norm_xtx_kernel(const float* __restrict__ coords,
                                       const float* __restrict__ icv,
                                       const float* __restrict__ ncvIn,
                                       const int* __restrict__ batch,
                                       const float* __restrict__ comcnt,
                                       float* __restrict__ ncvOut,
                                       float* __restrict__ xtx, int useIC) {
  int t = blockIdx.x * blockDim.x + threadIdx.x;
  int base = t * STRIP;
  if (base >= N_NODES) return;
  int end = base + STRIP; if (end > N_NODES) end = N_NODES;
  float s[6] = {0.f, 0.f, 0.f, 0.f, 0.f, 0.f};
  int cur = batch[base];
  for (int n = base; n < end; ++n) {
    int b = batch[n];
    if (b != cur) {
      #pragma unroll
      for (int i = 0; i < 6; ++i) { atomicAdd(&xtx[cur * 6 + i], s[i]); s[i] = 0.f; }
      cur = b;
    }
    float x, y, z;
    if (useIC) { float sc = icv[n]; x = sc * ncvIn[n * 3 + 0]; y = sc * ncvIn[n * 3 + 1]; z = sc * ncvIn[n * 3 + 2]; }
    else       { x = coords[n * 3 + 0]; y = coords[n * 3 + 1]; z = coords[n * 3 + 2]; }
    float cx = x - comcnt[b * 4 + 0];
    float cy = y - comcnt[b * 4 + 1];
    float cz = z - comcnt[b * 4 + 2];
    float rin = rsqrtf(cx * cx + cy * cy + cz * cz);
    ncvOut[n * 3 + 0] = cx * rin; ncvOut[n * 3 + 1] = cy * rin; ncvOut[n * 3 + 2] = cz * rin;
    s[0] += cx * cx; s[1] += cx * cy; s[2] += cx * cz;
    s[3] += cy * cy; s[4] += cy * cz; s[5] += cz * cz;
  }
  #pragma unroll
  for (int i = 0; i < 6; ++i) atomicAdd(&xtx[cur * 6 + i], s[i]);
}

__global__ void finalize_xtx_kernel(float* __restrict__ xtx, const float* __restrict__ comcnt) {
  int g = blockIdx.x * blockDim.x + threadIdx.x;
  if (g >= NG) return;
  float inv = 1.0f / fmaxf(comcnt[g * 4 + 3], 1.0f);
  #pragma unroll
  for (int i = 0; i < 6; ++i) xtx[g * 6 + i] *= inv;
}

// ---------------------------------------------------------------------------
// Analytic symmetric 3x3 eigendecomposition, eigenvalues ascending.
// Eigenvector signs are arbitrary; the downstream features are sign-invariant.
// ---------------------------------------------------------------------------
__device__ __forceinline__ void cross3(const float* u, const float* v, float* o) {
  o[0] = u[1] * v[2] - u[2] * v[1];
  o[1] = u[2] * v[0] - u[0] * v[2];
  o[2] = u[0] * v[1] - u[1] * v[0];
}
__device__ __forceinline__ float dot3(const float* u, const float* v) {
  return u[0] * v[0] + u[1] * v[1] + u[2] * v[2];
}
__device__ void sym3_eigvec(float a, float b, float c, float d, float e, float f,
                            float lam, int axis, float* out) {
  float r0[3] = {a - lam, b, c}, r1[3] = {b, d - lam, e}, r2[3] = {c, e, f - lam};
  float c0[3], c1[3], c2[3];
  cross3(r0, r1, c0); cross3(r0, r2, c1); cross3(r1, r2, c2);
  float n0 = dot3(c0, c0), n1 = dot3(c1, c1), n2 = dot3(c2, c2);
  const float* best = c0; float nb = n0;
  if (n1 > nb) { best = c1; nb = n1; }
  if (n2 > nb) { best = c2; nb = n2; }
  if (nb < 1e-30f) { out[0] = out[1] = out[2] = 0.f; out[axis] = 1.f; return; }
  float inv = rsqrtf(nb);
  out[0] = best[0] * inv; out[1] = best[1] * inv; out[2] = best[2] * inv;
}

__global__ void eigh3_kernel(const float* __restrict__ xtx, float* __restrict__ eig) {
  int g = blockIdx.x * blockDim.x + threadIdx.x;
  if (g >= NG) return;
  float a = xtx[g * 6 + 0], b = xtx[g * 6 + 1], c = xtx[g * 6 + 2];
  float d = xtx[g * 6 + 3], e = xtx[g * 6 + 4], f = xtx[g * 6 + 5];
  float q = (a + d + f) * (1.f / 3.f);
  float b00 = a - q, b11 = d - q, b22 = f - q;
  float p2 = b00 * b00 + b11 * b11 + b22 * b22 + 2.f * (b * b + c * c + e * e);
  float p = sqrtf(p2 * (1.f / 6.f));
  float l0, l1, l2;
  if (p < 1e-20f) { l0 = l1 = l2 = q; }
  else {
    float ip = 1.f / p;
    float B00 = b00 * ip, B11 = b11 * ip, B22 = b22 * ip;
    float B01 = b * ip, B02 = c * ip, B12 = e * ip;
    float det = B00 * (B11 * B22 - B12 * B12) - B01 * (B01 * B22 - B12 * B02)
              + B02 * (B01 * B12 - B11 * B02);
    float r = fminf(1.f, fmaxf(-1.f, 0.5f * det));
    float phi = acosf(r) * (1.f / 3.f);
    l2 = q + 2.f * p * __cosf(phi);
    l0 = q + 2.f * p * __cosf(phi + 2.0943951023931953f);  // + 2*pi/3
    l1 = 3.f * q - l0 - l2;
  }
  float v0[3], v1[3], v2[3];
  sym3_eigvec(a, b, c, d, e, f, l0, 0, v0);
  sym3_eigvec(a, b, c, d, e, f, l2, 2, v2);
  cross3(v2, v0, v1);
  float n1 = dot3(v1, v1);
  if (n1 < 1e-20f) sym3_eigvec(a, b, c, d, e, f, l1, 1, v1);
  else { float inv = rsqrtf(n1); v1[0] *= inv; v1[1] *= inv; v1[2] *= inv; }
  #pragma unroll
  for (int k = 0; k < 3; ++k) {
    eig[g * 9 + 0 + k] = v0[k];  // row 0 = smallest eigenvalue
    eig[g * 9 + 3 + k] = v1[k];
    eig[g * 9 + 6 + k] = v2[k];  // row 2 = largest
  }
}

// ---------------------------------------------------------------------------
// Segment start offsets (batch is sorted): atomicMin + backward fill.
// ---------------------------------------------------------------------------
__global__ void starts_min_kernel(const int* __restrict__ batch, int* __restrict__ starts) {
  int n = blockIdx.x * blockDim.x + threadIdx.x;
  if (n < N_NODES) atomicMin(&starts[batch[n]], n);
}
__global__ void starts_fix_kernel(int* __restrict__ starts) {
  if (threadIdx.x == 0 && blockIdx.x == 0) {
    starts[NG] = N_NODES;
    for (int g = NG - 1; g >= 0; --g)
      if (starts[g] > N_NODES) starts[g] = starts[g + 1];
  }
}

// ---------------------------------------------------------------------------
// Fused GEMM + bias + silu + (.@w2 + b2) using V_WMMA_F32_16X16X4_F32.
// X:[M,K] (K % 4 == 0), W:[K,192], out:[M].
// One wave owns a 16-row tile with 12 v8f accumulators (all 192 columns), so
// X and W are each read once per tile; the hidden layer never hits memory.
// A frag (16x4 f32, 2 VGPRs): lane l (m=l&15, kh=l>>4) holds K = kh*2+{0,1}.
// B frag (4x16, 2 VGPRs): vgpr j, lane l holds B[kh*2+j][l&15].
// C (16x16 f32, 8 VGPRs): vgpr r, lane l holds D[kh*8+r][l&15].
// ---------------------------------------------------------------------------
__global__ __launch_bounds__(256)
void mlp_silu_dot_wmma(const float* __restrict__ X, int M, int K,
                       const float* __restrict__ W, const float* __restrict__ bias,
                       const float* __restrict__ w2, const float* __restrict__ b2p,
                       float* __restrict__ out) {
  const int wave = threadIdx.x >> 5;
  const int lane = threadIdx.x & 31;
  const int mrow = lane & 15;
  const int khalf = lane >> 4;
  const int tileBase = (blockIdx.x * 8 + wave) * 16;
  if (tileBase >= M) return;           // wave-uniform: EXEC stays all-ones for WMMA
  int row = tileBase + mrow;
  if (row >= M) row = M - 1;           // clamp (reads only; writes are guarded)
  const float* xr = X + (size_t)row * K;

  v8f acc[12];
  #pragma unroll
  for (int nt = 0; nt < 12; ++nt) acc[nt] = (v8f){0.f, 0.f, 0.f, 0.f, 0.f, 0.f, 0.f, 0.f};

  for (int k0 = 0; k0 < K; k0 += 4) {
    v2f af;
    af.x = xr[k0 + khalf * 2 + 0];
    af.y = xr[k0 + khalf * 2 + 1];
    const float* w0 = W + (size_t)(k0 + khalf * 2) * CH;
    #pragma unroll
    for (int nt = 0; nt < 12; ++nt) {
      v2f bf;
      bf.x = w0[nt * 16 + mrow];
      bf.y = w0[CH + nt * 16 + mrow];
      acc[nt] = __builtin_amdgcn_wmma_f32_16x16x4_f32(
          false, af, false, bf, (short)0, acc[nt], false, false);
    }
  }

  float b2v = b2p[0];
  float rs[8];
  #pragma unroll
  for (int r = 0; r < 8; ++r) rs[r] = 0.f;
  #pragma unroll
  for (int nt = 0; nt < 12; ++nt) {
    int col = nt * 16 + mrow;
    float bb = bias[col], ww = w2[col];
    #pragma unroll
    for (int r = 0; r < 8; ++r) {
      float h = acc[nt][r] + bb;
      rs[r] += silu_f(h) * ww;
    }
  }
  // reduce over the 16 column-lanes inside each half-wave
  #pragma unroll
  for (int off = 8; off >= 1; off >>= 1) {
    #pragma unroll
    for (int r = 0; r < 8; ++r) rs[r] += __shfl_xor(rs[r], off, 32);
  }
  if (mrow == 0) {
    #pragma unroll
    for (int r = 0; r < 8; ++r) {
      int rr = tileBase + khalf * 8 + r;
      if (rr < M) out[rr] = rs[r] + b2v;
    }
  }
}

// ---------------------------------------------------------------------------
// One workgroup per graph: stream that graph's nodes once, accumulate
// pos/neg x (spatial,inter) sums [4][192][3] in LDS (thread==channel, no
// atomics), then mean -> abs -> Wd down-projection and feature assembly.
// ---------------------------------------------------------------------------
__global__ __launch_bounds__(192)
void proj_kernel(const float* __restrict__ feats,
                 const float* __restrict__ ncv,
                 const float* __restrict__ incv,
                 const float* __restrict__ eig_s,
                 const float* __restrict__ eig_i,
                 const int* __restrict__ starts,
                 const float* __restrict__ comcnt,
                 const float* __restrict__ Wd,      // [CH, CDN]
                 float* __restrict__ featbuf, int off) {
  __shared__ float acc[4][CH][3];
  __shared__ float down[4][CDN][3];
  __shared__ float es[9], ei[9];
  int g = blockIdx.x, tid = threadIdx.x;
  if (tid < 9) { es[tid] = eig_s[g * 9 + tid]; ei[tid] = eig_i[g * 9 + tid]; }
  for (int i = tid; i < 4 * CH * 3; i += blockDim.x) (&acc[0][0][0])[i] = 0.f;
  __syncthreads();

  int s0 = starts[g], s1 = starts[g + 1];
  int c = tid;  // channel 0..191
  for (int n = s0; n < s1; ++n) {
    if (n + 1 < s1) __builtin_prefetch(&feats[(size_t)(n + 1) * CH + c], 0, 0);
    float nx = ncv[n * 3 + 0], ny = ncv[n * 3 + 1], nz = ncv[n * 3 + 2];
    float jx = incv[n * 3 + 0], jy = incv[n * 3 + 1], jz = incv[n * 3 + 2];
    float v = feats[(size_t)n * CH + c];
    #pragma unroll
    for (int t = 0; t < 3; ++t) {
      float cs = nx * es[t * 3] + ny * es[t * 3 + 1] + nz * es[t * 3 + 2];
      float ci = jx * ei[t * 3] + jy * ei[t * 3 + 1] + jz * ei[t * 3 + 2];
      float ps = cs * v, pi = ci * v;
      if (cs >= 0.f) acc[0][c][t] += ps; else acc[1][c][t] += ps;
      if (ci >= 0.f) acc[2][c][t] += pi; else acc[3][c][t] += pi;
    }
  }
  __syncthreads();

  float invc = 1.0f / fmaxf(comcnt[g * 4 + 3], 1.0f);
  // lin_down: 4 buffers x CDN x 3 = 384 dots of length 192 (abs before Wd)
  for (int idx = tid; idx < 4 * CDN * 3; idx += blockDim.x) {
    int qb = idx / (CDN * 3);
    int rem = idx % (CDN * 3);
    int dd = rem / 3, t = rem % 3;
    float sum = 0.f;
    for (int cc = 0; cc < CH; ++cc) sum += fabsf(acc[qb][cc][t]) * Wd[cc * CDN + dd];
    down[qb][dd][t] = sum * invc;
  }
  __syncthreads();

  float* fb = featbuf + (size_t)g * KPAD + off;
  for (int j = tid; j < CH; j += blockDim.x) {   // 6*CDN == 192 outputs per block type
    int grp = j / CDN, dd = j % CDN;
    float vs, vi;
    if (grp < 3) { vs = down[0][dd][grp] + down[1][dd][grp];
                   vi = down[2][dd][grp] + down[3][dd][grp]; }
    else         { vs = down[0][dd][grp - 3] * down[1][dd][grp - 3];
                   vi = down[2][dd][grp - 3] * down[3][dd][grp - 3]; }
    fb[j] = vs;
    fb[CH + j] = vi;
  }
  if (tid < 9) {
    int i = tid / 3, jj = tid % 3;
    float dv = fabsf(es[i * 3] * ei[jj * 3] + es[i * 3 + 1] * ei[jj * 3 + 1]
                   + es[i * 3 + 2] * ei[jj * 3 + 2]);
    fb[2 * CH + tid] = dv;
  }
}

// Zero-padded copy of H1 [786,192] -> [KPAD,192]
__global__ void build_h1pad_kernel(const float* __restrict__ H1, float* __restrict__ h1pad) {
  int i = blockIdx.x * blockDim.x + threadIdx.x;
  if (i >= KPAD * CH) return;
  int k = i / CH, n = i % CH;
  h1pad[i] = (k < 2 * FEAT1) ? H1[k * CH + n] : 0.f;
}

// ---------------------------------------------------------------------------
extern "C" void kernel_launch(void* const* d_in, const int* in_sizes, int n_in,
                              void* d_out, int out_size, void* d_ws, size_t ws_size,
                              hipStream_t stream) {
  const float* feats_a  = (const float*)d_in[0];
  const float* coords_a = (const float*)d_in[1];
  const float* feats_b  = (const float*)d_in[2];
  const float* coords_b = (const float*)d_in[3];
  const int*   batch_a  = (const int*)d_in[4];
  const int*   batch_b  = (const int*)d_in[5];
  // d_in[6] = num_graphs (compile-time NG)
  const float* W1  = (const float*)d_in[7];
  const float* b1  = (const float*)d_in[8];
  const float* W2  = (const float*)d_in[9];
  const float* b2  = (const float*)d_in[10];
  const float* Wd  = (const float*)d_in[11];
  const float* H1  = (const float*)d_in[12];
  const float* bh1 = (const float*)d_in[13];
  const float* H2  = (const float*)d_in[14];
  const float* bh2 = (const float*)d_in[15];
  float* out = (float*)d_out;

  // workspace layout (floats)
  float* ws      = (float*)d_ws;
  float* ncv     = ws;                                // N*3
  float* incv    = ncv  + (size_t)N_NODES * 3;        // N*3
  float* icv     = incv + (size_t)N_NODES * 3;        // N
  float* comcnt  = icv  + N_NODES;                    // G*4
  float* xtx_s   = comcnt + NG * 4;                   // G*6
  float* xtx_i   = xtx_s + NG * 6;                    // G*6
  float* eig_s   = xtx_i + NG * 6;                    // G*9
  float* eig_i   = eig_s + NG * 9;                    // G*9
  int*   starts  = (int*)(eig_i + NG * 9);            // G+1 (+pad)
  float* featbuf = (float*)(starts + NG + 4);         // G*KPAD
  float* h1pad   = featbuf + (size_t)NG * KPAD;       // KPAD*192

  hipMemsetAsync(featbuf, 0, (size_t)NG * KPAD * sizeof(float), stream);
  { int tot = KPAD * CH;
    build_h1pad_kernel<<<(tot + 255) / 256, 256, 0, stream>>>(H1, h1pad); }

  const int segBlocks  = (N_NODES + 256 * STRIP - 1) / (256 * STRIP);
  const int nodeBlocks = (N_NODES + 255) / 256;

  for (int e = 0; e < 2; ++e) {
    const float* feats  = e ? feats_b  : feats_a;
    const float* coords = e ? coords_b : coords_a;
    const int*   batch  = e ? batch_b  : batch_a;
    const int    off    = e ? FEAT1 : 0;

    // spatial eigen basis
    hipMemsetAsync(comcnt, 0, NG * 4 * sizeof(float), stream);
    hipMemsetAsync(xtx_s, 0, NG * 6 * sizeof(float), stream);
    seg_sum_vec3_kernel<<<segBlocks, 256, 0, stream>>>(coords, nullptr, nullptr, batch, comcnt, 0);
    finalize_com_kernel<<<2, 256, 0, stream>>>(comcnt);
    center_norm_xtx_kernel<<<segBlocks, 256, 0, stream>>>(coords, nullptr, nullptr, batch, comcnt,
                                                          ncv, xtx_s, 0);
    finalize_xtx_kernel<<<2, 256, 0, stream>>>(xtx_s, comcnt);
    eigh3_kernel<<<2, 256, 0, stream>>>(xtx_s, eig_s);

    // interaction coefficients MLP (WMMA f32 GEMM, fused silu + W2 dot)
    { int tiles = (N_NODES + 15) / 16, blocks = (tiles + 7) / 8;
      mlp_silu_dot_wmma<<<blocks, 256, 0, stream>>>(feats, N_NODES, CH, W1, b1, W2, b2, icv); }

    // interaction eigen basis (vectors = ic * ncv)
    hipMemsetAsync(comcnt, 0, NG * 4 * sizeof(float), stream);
    hipMemsetAsync(xtx_i, 0, NG * 6 * sizeof(float), stream);
    seg_sum_vec3_kernel<<<segBlocks, 256, 0, stream>>>(nullptr, icv, ncv, batch, comcnt, 1);
    finalize_com_kernel<<<2, 256, 0, stream>>>(comcnt);
    center_norm_xtx_kernel<<<segBlocks, 256, 0, stream>>>(nullptr, icv, ncv, batch, comcnt,
                                                          incv, xtx_i, 1);
    finalize_xtx_kernel<<<2, 256, 0, stream>>>(xtx_i, comcnt);
    eigh3_kernel<<<2, 256, 0, stream>>>(xtx_i, eig_i);

    // segment ranges
    hipMemsetAsync(starts, 0x7F, (NG + 1) * sizeof(int), stream);
    starts_min_kernel<<<nodeBlocks, 256, 0, stream>>>(batch, starts);
    starts_fix_kernel<<<1, 32, 0, stream>>>(starts);

    // fused spatial+interaction projection pooling + lin_down + dots
    proj_kernel<<<NG, 192, 0, stream>>>(feats, ncv, incv, eig_s, eig_i, starts, comcnt,
                                        Wd, featbuf, off);
  }

  // head MLP on [G, KPAD] padded features (same WMMA kernel)
  { int tiles = (NG + 15) / 16, blocks = (tiles + 7) / 8;
    mlp_silu_dot_wmma<<<blocks, 256, 0, stream>>>(featbuf, NG, KPAD, h1pad, bh1, H2, bh2, out); }
}